// MeshNN_1D_26010321944831
// MI455X (gfx1250) — compile-verified
//
#include <hip/hip_runtime.h>
#include <hip/hip_bf16.h>
#include <stdint.h>

#define TPB 256
#define EPT 4                    // elements per thread
#define EPB (TPB * EPT)          // 1024 elements per block -> 8KB conn tile

typedef __attribute__((ext_vector_type(4))) unsigned int v4u;
typedef __attribute__((ext_vector_type(8))) int          v8i;
typedef __attribute__((ext_vector_type(4))) int          v4i;
typedef __attribute__((ext_vector_type(4))) float        f4;

// Gauss-Legendre tables matching the reference _gauss() ordering exactly.
template<int G> struct GaussTab;
template<> struct GaussTab<1> {
  static constexpr float xi[1] = {0.0f};
  static constexpr float w [1] = {2.0f};
};
template<> struct GaussTab<2> {
  static constexpr float xi[2] = {-0.57735026918962576f, 0.57735026918962576f};
  static constexpr float w [2] = {1.0f, 1.0f};
};
template<> struct GaussTab<3> {
  static constexpr float xi[3] = {-0.77459666924148338f, 0.0f, 0.77459666924148338f};
  static constexpr float w [3] = {0.55555555555555556f, 0.88888888888888889f, 0.55555555555555556f};
};
template<> struct GaussTab<4> {
  static constexpr float xi[4] = {-0.86113631159405258f, -0.33998104358485626f,
                                   0.33998104358485626f,  0.86113631159405258f};
  static constexpr float w [4] = {0.34785484513745386f, 0.65214515486254614f,
                                  0.65214515486254614f, 0.34785484513745386f};
};
template<> struct GaussTab<5> {
  static constexpr float xi[5] = {0.0f, -0.53846931010568309f, 0.53846931010568309f,
                                        -0.90617984593866399f, 0.90617984593866399f};
  static constexpr float w [5] = {0.56888888888888889f, 0.47862867049936647f, 0.47862867049936647f,
                                  0.23692688505618909f, 0.23692688505618909f};
};

template<int G>
__global__ __launch_bounds__(TPB) void fe_eval_kernel(
    const float* __restrict__ coords,
    const float* __restrict__ vals,
    const int*   __restrict__ conn,      // [E][2], 1-based node ids
    float* __restrict__ outI,            // interpol [E*G]
    float* __restrict__ outX,            // x_g      [E*G]
    float* __restrict__ outD,            // detJ_w   [E*G]
    int E)
{
  __shared__ __align__(16) int lconn[EPB * 2];   // 8KB connectivity tile

  const int tid     = (int)threadIdx.x;
  const int blockE0 = (int)blockIdx.x * EPB;
  const int e0      = blockE0 + tid * EPT;

  int4 cab, ccd;

#if defined(__HIP_DEVICE_COMPILE__) && \
    __has_builtin(__builtin_amdgcn_tensor_load_to_lds) && \
    __has_builtin(__builtin_amdgcn_s_wait_tensorcnt)
  // --- TDM path: wave 0 DMAs this block's conn tile Global->LDS ---
  if (tid < 32) {
    uint64_t gaddr  = (uint64_t)(uintptr_t)(conn + (size_t)blockE0 * 2);
    uint32_t ldsoff = (uint32_t)(uintptr_t)(&lconn[0]);
    // Remaining units from tile start: tail blocks read zero-fill instead of OOB.
    uint32_t remain = (uint32_t)((uint64_t)E * 2u - (uint64_t)blockE0 * 2u);

    v4u g0; v8i g1; v4i g2; v4i g3;
    // D# group 0: count=1, lds_addr, global_addr[56:0], type=2 ("image")
    g0[0] = 1u;
    g0[1] = ldsoff;
    g0[2] = (unsigned int)gaddr;
    g0[3] = (unsigned int)((gaddr >> 32) & 0x1FFFFFFu) | (2u << 30);
    // D# group 1: data_size=4B (code 2); tensor_dim0=remain; tile_dim0=2048 units (1-D tile)
    g1[0] = (int)(2u << 16);                    // workgroup_mask=0, data_size=2 (4B)
    g1[1] = (int)((remain & 0xFFFFu) << 16);    // tensor_dim0[15:0]  -> bits 63:48
    g1[2] = (int)((remain >> 16) & 0xFFFFu);    // tensor_dim0[31:16] -> bits 79:64
    g1[3] = (int)((unsigned)(EPB * 2) << 16);   // tile_dim0 -> bits 127:112
    g1[4] = 0;                                  // tile_dim1=0, tile_dim2=0 (unused)
    g1[5] = (int)(EPB * 2);                     // tensor_dim0_stride (unused for 1-D)
    g1[6] = 0; g1[7] = 0;
    g2 = (v4i)0; g3 = (v4i)0;
  #if __clang_major__ >= 23
    v8i g4 = (v8i)0;
    __builtin_amdgcn_tensor_load_to_lds(g0, g1, g2, g3, g4, 0);
  #else
    __builtin_amdgcn_tensor_load_to_lds(g0, g1, g2, g3, 0);
  #endif
    __builtin_amdgcn_s_wait_tensorcnt(0);
  }
  __syncthreads();
  {
    const int4* cp = reinterpret_cast<const int4*>(&lconn[tid * (EPT * 2)]);
    cab = cp[0];
    ccd = cp[1];
  }
#else
  // --- fallback: direct coalesced b128 loads of connectivity ---
  if (e0 + EPT <= E) {
    const int4* gp = reinterpret_cast<const int4*>(conn + (size_t)e0 * 2);
    cab = gp[0];
    ccd = gp[1];
  } else {
    int tmp[8] = {1, 1, 1, 1, 1, 1, 1, 1};
    for (int k = 0; k < EPT; ++k) {
      if (e0 + k < E) {
        tmp[2 * k]     = conn[(size_t)(e0 + k) * 2];
        tmp[2 * k + 1] = conn[(size_t)(e0 + k) * 2 + 1];
      }
    }
    cab = make_int4(tmp[0], tmp[1], tmp[2], tmp[3]);
    ccd = make_int4(tmp[4], tmp[5], tmp[6], tmp[7]);
  }
  (void)lconn;
#endif

  const int idx[EPT * 2] = {cab.x, cab.y, cab.z, cab.w, ccd.x, ccd.y, ccd.z, ccd.w};

  // Warm L2 for the next tile of node data (speculative; dropped if invalid).
  __builtin_prefetch(coords + idx[0] + EPB, 0, 0);
  __builtin_prefetch(vals   + idx[0] + EPB, 0, 0);

  alignas(16) float bufI[EPT * G];
  alignas(16) float bufX[EPT * G];
  alignas(16) float bufD[EPT * G];

  const bool full = (e0 + EPT <= E);

#pragma unroll
  for (int k = 0; k < EPT; ++k) {
    const bool valid = (e0 + k) < E;
    const int i1 = valid ? (idx[2 * k]     - 1) : 0;
    const int i2 = valid ? (idx[2 * k + 1] - 1) : 0;
    const float x1 = coords[i1];
    const float x2 = coords[i2];
    const float v1 = vals[i1];
    const float v2 = vals[i2];
    const float dx  = x2 - x1;
    const float inv = 1.0f / dx;
#pragma unroll
    for (int g = 0; g < G; ++g) {
      const float t  = (GaussTab<G>::xi[g] + 1.0f) * dx * 0.5f;  // (xi+1)(x2-x1)/2
      const float xg = x1 + t;                                   // physical Gauss pt
      const float r  = 2.0f * (xg - x1) * inv - 1.0f;            // back to ref coords
      const float n0 = 0.5f - 0.5f * r;
      const float n1 = 0.5f + 0.5f * r;
      bufI[k * G + g] = n0 * v1 + n1 * v2;
      bufX[k * G + g] = xg;
      bufD[k * G + g] = dx * 0.5f * GaussTab<G>::w[g];
    }
  }

  const size_t base = (size_t)e0 * (size_t)G;
  if (full) {
    // EPT*G is a multiple of 4 for G=1..5; base is 16B-aligned (e0 % 4 == 0).
    constexpr int NV = (EPT * G) / 4;
    const f4* sI = reinterpret_cast<const f4*>(bufI);
    const f4* sX = reinterpret_cast<const f4*>(bufX);
    const f4* sD = reinterpret_cast<const f4*>(bufD);
    f4* dI = reinterpret_cast<f4*>(outI + base);
    f4* dX = reinterpret_cast<f4*>(outX + base);
    f4* dD = reinterpret_cast<f4*>(outD + base);
#pragma unroll
    for (int j = 0; j < NV; ++j) {
      __builtin_nontemporal_store(sI[j], dI + j);   // streaming: keep L2 for node data
      __builtin_nontemporal_store(sX[j], dX + j);
      __builtin_nontemporal_store(sD[j], dD + j);
    }
  } else {
    for (int k = 0; k < EPT; ++k) {
      if (e0 + k >= E) break;
      for (int g = 0; g < G; ++g) {
        outI[base + k * G + g] = bufI[k * G + g];
        outX[base + k * G + g] = bufX[k * G + g];
        outD[base + k * G + g] = bufD[k * G + g];
      }
    }
  }
}

extern "C" void kernel_launch(void* const* d_in, const int* in_sizes, int n_in,
                              void* d_out, int out_size, void* d_ws, size_t ws_size,
                              hipStream_t stream) {
  (void)n_in; (void)d_ws; (void)ws_size;
  const float* coords = (const float*)d_in[0];
  const float* vals   = (const float*)d_in[1];
  const int*   conn   = (const int*)d_in[2];   // [E][2] int32

  const int E = in_sizes[2] / 2;
  int G = 3;
  if (E > 0) {
    long long g = (long long)out_size / (3LL * (long long)E);
    if (g >= 1 && g <= 5) G = (int)g;
  }

  float* outI = (float*)d_out;
  float* outX = outI + (size_t)E * (size_t)G;
  float* outD = outX + (size_t)E * (size_t)G;

  const int blocks = (E + EPB - 1) / EPB;

  switch (G) {
    case 1: fe_eval_kernel<1><<<blocks, TPB, 0, stream>>>(coords, vals, conn, outI, outX, outD, E); break;
    case 2: fe_eval_kernel<2><<<blocks, TPB, 0, stream>>>(coords, vals, conn, outI, outX, outD, E); break;
    case 3: fe_eval_kernel<3><<<blocks, TPB, 0, stream>>>(coords, vals, conn, outI, outX, outD, E); break;
    case 4: fe_eval_kernel<4><<<blocks, TPB, 0, stream>>>(coords, vals, conn, outI, outX, outD, E); break;
    case 5: fe_eval_kernel<5><<<blocks, TPB, 0, stream>>>(coords, vals, conn, outI, outX, outD, E); break;
  }
}